// GraphNN_38577396253199
// MI455X (gfx1250) — compile-verified
//
#include <hip/hip_runtime.h>

typedef float v2f __attribute__((ext_vector_type(2)));
typedef float v8f __attribute__((ext_vector_type(8)));

#define DDIM 128

// ---------------- degree / normalization ----------------

__global__ void k_deg_init(float* __restrict__ deg, int n) {
    int i = blockIdx.x * blockDim.x + threadIdx.x;
    if (i < n) deg[i] = 1.0f;  // self-loop weight
}

__global__ void k_deg_scatter(const int* __restrict__ dst,
                              const float* __restrict__ ew,
                              float* __restrict__ deg, int E) {
    int e = blockIdx.x * blockDim.x + threadIdx.x;
    if (e < E) {
        (void)__hip_atomic_fetch_add(&deg[dst[e]], ew[e],
                                     __ATOMIC_RELAXED, __HIP_MEMORY_SCOPE_AGENT);
    }
}

__global__ void k_dinv(float* __restrict__ deg, int n) {
    int i = blockIdx.x * blockDim.x + threadIdx.x;
    if (i < n) {
        float d = deg[i];
        deg[i] = (d > 0.0f) ? rsqrtf(d) : 0.0f;
    }
}

// ---------------- x = emb @ W via V_WMMA_F32_16X16X4_F32 ----------------
// Each wave computes one 16-row block across all 128 output columns
// (8 accumulator tiles), chaining 32 k-step WMMAs per tile.
// A layout: lane L, vgpr v -> A[m = L%16][k = (L/16)*2 + v]
// B layout: lane L, vgpr v -> B[k = (L/16)*2 + v][n = L%16]
// C layout: vgpr r, lane L -> C[m = r + (L/16)*8][n = L%16]

__global__ void k_gemm_wmma(const float* __restrict__ emb,
                            const float* __restrict__ W,
                            float* __restrict__ x, int nTilesM) {
    int lane  = threadIdx.x & 31;
    int wave  = threadIdx.x >> 5;
    int mTile = blockIdx.x * (blockDim.x >> 5) + wave;
    if (mTile >= nTilesM) return;   // wave-uniform branch

    int mlo   = lane & 15;
    int khalf = (lane >> 4) << 1;   // 0 or 2

    v8f acc[8];
#pragma unroll
    for (int nt = 0; nt < 8; ++nt)
        acc[nt] = (v8f){0.f, 0.f, 0.f, 0.f, 0.f, 0.f, 0.f, 0.f};

    const float* arow = emb + (size_t)(mTile * 16 + mlo) * DDIM;

    for (int kc = 0; kc < DDIM; kc += 4) {
        v2f a;
        a.x = arow[kc + khalf];
        a.y = arow[kc + khalf + 1];
        const float* bp = W + (size_t)(kc + khalf) * DDIM + mlo;
#pragma unroll
        for (int nt = 0; nt < 8; ++nt) {
            v2f b;
            b.x = bp[nt * 16];
            b.y = bp[nt * 16 + DDIM];
            acc[nt] = __builtin_amdgcn_wmma_f32_16x16x4_f32(
                false, a, false, b, (short)0, acc[nt], false, false);
        }
    }

    int crow = mTile * 16 + ((lane >> 4) << 3);
#pragma unroll
    for (int nt = 0; nt < 8; ++nt) {
        float* cp = x + (size_t)crow * DDIM + nt * 16 + mlo;
#pragma unroll
        for (int r = 0; r < 8; ++r)
            cp[(size_t)r * DDIM] = acc[nt][r];
    }
}

// Scalar tail for rows not covered by 16-row tiles (not needed for N=100000).
__global__ void k_gemm_tail(const float* __restrict__ emb,
                            const float* __restrict__ W,
                            float* __restrict__ x, int startRow, int n) {
    int row = startRow + blockIdx.x;
    if (row >= n) return;
    int d = threadIdx.x;
    float acc = 0.f;
    for (int k = 0; k < DDIM; ++k)
        acc = fmaf(emb[(size_t)row * DDIM + k], W[(size_t)k * DDIM + d], acc);
    x[(size_t)row * DDIM + d] = acc;
}

// ---------------- out init: self-loop contribution + bias ----------------

__global__ void k_self(const float* __restrict__ x,
                       const float* __restrict__ dinv,
                       const float* __restrict__ b,
                       float* __restrict__ out) {
    int i = blockIdx.x;
    int d = threadIdx.x;
    float di = dinv[i];
    out[(size_t)i * DDIM + d] = x[(size_t)i * DDIM + d] * (di * di) + b[d];
}

// ---------------- weighted edge scatter-add ----------------
// One 128-thread block per edge: coalesced 512B row gather, f32 atomics.

__global__ void k_edge(const int* __restrict__ src,
                       const int* __restrict__ dst,
                       const float* __restrict__ ew,
                       const float* __restrict__ x,
                       const float* __restrict__ dinv,
                       float* __restrict__ out) {
    int e = blockIdx.x;
    int d = threadIdx.x;
    int s = src[e];
    int t = dst[e];
    float c = dinv[s] * ew[e] * dinv[t];
    float v = x[(size_t)s * DDIM + d] * c;
    (void)__hip_atomic_fetch_add(&out[(size_t)t * DDIM + d], v,
                                 __ATOMIC_RELAXED, __HIP_MEMORY_SCOPE_AGENT);
}

// ---------------- launcher ----------------

extern "C" void kernel_launch(void* const* d_in, const int* in_sizes, int n_in,
                              void* d_out, int out_size, void* d_ws, size_t ws_size,
                              hipStream_t stream) {
    const float* emb = (const float*)d_in[0];
    const float* W   = (const float*)d_in[1];
    const float* b   = (const float*)d_in[2];
    const int*   ei  = (const int*)d_in[3];   // [2, E]: src then dst
    const float* ew  = (const float*)d_in[4];

    const int n = in_sizes[0] / DDIM;  // 100000
    const int E = in_sizes[4];         // 1600000

    float* x   = (float*)d_ws;                 // [n, 128]
    float* deg = x + (size_t)n * DDIM;         // [n]  (becomes dinv in place)
    float* out = (float*)d_out;

    const int* src = ei;
    const int* dst = ei + E;

    // 1) degree with self-loops -> dinv (in place)
    k_deg_init<<<(n + 255) / 256, 256, 0, stream>>>(deg, n);
    k_deg_scatter<<<(E + 255) / 256, 256, 0, stream>>>(dst, ew, deg, E);
    k_dinv<<<(n + 255) / 256, 256, 0, stream>>>(deg, n);

    // 2) x = emb @ W  (WMMA f32 16x16x4, one 16-row block per wave)
    const int nTilesM = n / 16;
    const int wavesPerBlock = 4;                 // 128 threads
    const int nBlocks = (nTilesM + wavesPerBlock - 1) / wavesPerBlock;
    k_gemm_wmma<<<nBlocks, wavesPerBlock * 32, 0, stream>>>(emb, W, x, nTilesM);
    const int rem = n - nTilesM * 16;
    if (rem > 0)
        k_gemm_tail<<<rem, DDIM, 0, stream>>>(emb, W, x, nTilesM * 16, n);

    // 3) out = x * dinv^2 + bias (initializes d_out)
    k_self<<<n, DDIM, 0, stream>>>(x, deg, b, out);

    // 4) out[dst] += x[src] * (dinv[src]*ew*dinv[dst])
    k_edge<<<E, DDIM, 0, stream>>>(src, dst, ew, x, deg, out);
}